// GAT_56281251447436
// MI455X (gfx1250) — compile-verified
//
#include <hip/hip_runtime.h>
#include <hip/hip_bf16.h>

typedef float v2f __attribute__((ext_vector_type(2)));
typedef float v8f __attribute__((ext_vector_type(8)));

#define F_IN 512
#define A0   8     // layer0 attention units (8 heads, dh=1)
#define U0   64    // layer0 value units
#define NC   64    // layer1 classes

#if __has_builtin(__builtin_amdgcn_global_load_async_to_lds_b128) && \
    __has_builtin(__builtin_amdgcn_s_wait_asynccnt)
#define USE_ASYNC_LDS 1
#endif

static __host__ __device__ inline int ceil_div(int a, int b) { return (a + b - 1) / b; }

#ifdef USE_ASYNC_LDS
typedef int v4i_vs __attribute__((vector_size(4 * sizeof(int))));
typedef __attribute__((address_space(1))) v4i_vs* gas_v4i_ptr;
typedef __attribute__((address_space(3))) v4i_vs* lds_v4i_ptr;

__device__ __forceinline__ void async_copy16(const float* gsrc, float* ldst)
{
    __builtin_amdgcn_global_load_async_to_lds_b128(
        (gas_v4i_ptr)(void*)gsrc,
        (lds_v4i_ptr)(void*)ldst, 0, 0);
}
#endif

// ---------------------------------------------------------------------------
// Layer 0 projections: one block = 16 node rows. 5 waves:
//   wave 0  -> [Q|K] = relu(X @ [Wq|Wk] + [bq|bk])   (16 cols)
//   wave1-4 -> V = X @ Wv                             (64 cols)
// X tile (16x512 f32) staged in LDS, pitch 516 (conflict-free).
// ---------------------------------------------------------------------------
__global__ __launch_bounds__(160)
void gat_proj0_kernel(const float* __restrict__ x,
                      const float* __restrict__ Wq, const float* __restrict__ bq,
                      const float* __restrict__ Wk, const float* __restrict__ bk,
                      const float* __restrict__ Wv,
                      float* __restrict__ Q, float* __restrict__ K,
                      float* __restrict__ V, int n)
{
    __shared__ float lds[16 * 516];
    const int row0 = blockIdx.x * 16;
    const int tid  = threadIdx.x;

    // cooperative, coalesced tile fill (16 rows x 128 float4)
    for (int idx = tid; idx < 16 * 128; idx += 160) {
        int r  = idx >> 7;
        int c4 = idx & 127;
        int rr = row0 + r; if (rr >= n) rr = n - 1;
#ifdef USE_ASYNC_LDS
        async_copy16(x + (size_t)rr * F_IN + c4 * 4, &lds[r * 516 + c4 * 4]);
#else
        float4 val = ((const float4*)(x + (size_t)rr * F_IN))[c4];
        *(float4*)&lds[r * 516 + c4 * 4] = val;
#endif
    }
#ifdef USE_ASYNC_LDS
    __builtin_amdgcn_s_wait_asynccnt(0);
#endif
    __syncthreads();

    const int wave  = tid >> 5;
    const int lane  = tid & 31;
    const int mrow  = lane & 15;
    const int khalf = (lane >> 4) * 2;     // lanes 16-31 hold K+2,K+3
    const int ncol  = lane & 15;
    const bool full = (row0 + 16 <= n);

    v8f c = {0.f, 0.f, 0.f, 0.f, 0.f, 0.f, 0.f, 0.f};

    if (wave == 0) {
        // fused Q|K: cols 0-7 from Wq, 8-15 from Wk (per-lane pointer select)
        const float* Wb = (ncol < 8) ? Wq : Wk;
        const int nc8 = ncol & 7;
        for (int k0 = 0; k0 < F_IN; k0 += 4) {
            v2f a = *(const v2f*)&lds[mrow * 516 + k0 + khalf];
            v2f b;
            b.x = Wb[(k0 + khalf)     * A0 + nc8];
            b.y = Wb[(k0 + khalf + 1) * A0 + nc8];
            c = __builtin_amdgcn_wmma_f32_16x16x4_f32(false, a, false, b,
                                                      (short)0, c, false, false);
        }
        const float bias = (ncol < 8) ? bq[nc8] : bk[nc8];
        float* outp = (ncol < 8) ? Q : K;
        if (full) {
            #pragma unroll
            for (int i = 0; i < 8; i++) {
                int node = row0 + i + ((lane >> 4) << 3);
                outp[(size_t)node * A0 + nc8] = fmaxf(c[i] + bias, 0.f);
            }
        } else {
            #pragma unroll
            for (int i = 0; i < 8; i++) {
                int node = row0 + i + ((lane >> 4) << 3);
                if (node < n) outp[(size_t)node * A0 + nc8] = fmaxf(c[i] + bias, 0.f);
            }
        }
    } else {
        const int cb = (wave - 1) * 16;
        for (int k0 = 0; k0 < F_IN; k0 += 4) {
            v2f a = *(const v2f*)&lds[mrow * 516 + k0 + khalf];
            v2f b;
            b.x = Wv[(k0 + khalf)     * U0 + cb + ncol];
            b.y = Wv[(k0 + khalf + 1) * U0 + cb + ncol];
            c = __builtin_amdgcn_wmma_f32_16x16x4_f32(false, a, false, b,
                                                      (short)0, c, false, false);
        }
        if (full) {
            #pragma unroll
            for (int i = 0; i < 8; i++) {
                int node = row0 + i + ((lane >> 4) << 3);
                V[(size_t)node * U0 + cb + ncol] = c[i];
            }
        } else {
            #pragma unroll
            for (int i = 0; i < 8; i++) {
                int node = row0 + i + ((lane >> 4) << 3);
                if (node < n) V[(size_t)node * U0 + cb + ncol] = c[i];
            }
        }
    }
}

// ---------------------------------------------------------------------------
// Layer 1 V projection: V1 = h @ Wv1 (64x64). 4 waves, 16-col slices.
// ---------------------------------------------------------------------------
__global__ __launch_bounds__(128)
void gat_proj1_kernel(const float* __restrict__ h, const float* __restrict__ Wv,
                      float* __restrict__ V, int n)
{
    __shared__ float lds[16 * 68];
    const int row0 = blockIdx.x * 16;
    const int tid  = threadIdx.x;

    for (int idx = tid; idx < 16 * 16; idx += 128) {
        int r  = idx >> 4;
        int c4 = idx & 15;
        int rr = row0 + r; if (rr >= n) rr = n - 1;
#ifdef USE_ASYNC_LDS
        async_copy16(h + (size_t)rr * U0 + c4 * 4, &lds[r * 68 + c4 * 4]);
#else
        float4 val = ((const float4*)(h + (size_t)rr * U0))[c4];
        *(float4*)&lds[r * 68 + c4 * 4] = val;
#endif
    }
#ifdef USE_ASYNC_LDS
    __builtin_amdgcn_s_wait_asynccnt(0);
#endif
    __syncthreads();

    const int wave  = tid >> 5;
    const int lane  = tid & 31;
    const int mrow  = lane & 15;
    const int khalf = (lane >> 4) * 2;
    const int ncol  = lane & 15;
    const int cb    = wave * 16;
    const bool full = (row0 + 16 <= n);

    v8f c = {0.f, 0.f, 0.f, 0.f, 0.f, 0.f, 0.f, 0.f};
    for (int k0 = 0; k0 < U0; k0 += 4) {
        v2f a = *(const v2f*)&lds[mrow * 68 + k0 + khalf];
        v2f b;
        b.x = Wv[(k0 + khalf)     * NC + cb + ncol];
        b.y = Wv[(k0 + khalf + 1) * NC + cb + ncol];
        c = __builtin_amdgcn_wmma_f32_16x16x4_f32(false, a, false, b,
                                                  (short)0, c, false, false);
    }
    if (full) {
        #pragma unroll
        for (int i = 0; i < 8; i++) {
            int node = row0 + i + ((lane >> 4) << 3);
            V[(size_t)node * NC + cb + ncol] = c[i];
        }
    } else {
        #pragma unroll
        for (int i = 0; i < 8; i++) {
            int node = row0 + i + ((lane >> 4) << 3);
            if (node < n) V[(size_t)node * NC + cb + ncol] = c[i];
        }
    }
}

// ---------------------------------------------------------------------------
// Layer 0 edge-phase kernels. qk >= 0 always (relu'd Q,K with dh=1), so
// float max == uint max on bit patterns. Self loops handled analytically.
// ---------------------------------------------------------------------------
__global__ __launch_bounds__(256)
void gat_init0_kernel(const float* __restrict__ Q, const float* __restrict__ K,
                      float* __restrict__ m, float* __restrict__ z,
                      float* __restrict__ acc, int total /* n*64 */)
{
    int idx = blockIdx.x * blockDim.x + threadIdx.x;
    if (idx >= total) return;
    acc[idx] = 0.f;
    if ((idx & 63) < A0) {
        int i = idx >> 6, hh = idx & 7;
        float qk = Q[i * A0 + hh] * K[i * A0 + hh];   // self-loop score
        m[i * A0 + hh] = qk;
        z[i * A0 + hh] = 0.f;
    }
}

__global__ __launch_bounds__(256)
void gat_edge0_max_kernel(const int* __restrict__ row, const int* __restrict__ col,
                          const float* __restrict__ Q, const float* __restrict__ K,
                          float* __restrict__ m, int total /* E*8 */)
{
    int idx = blockIdx.x * blockDim.x + threadIdx.x;
    if (idx >= total) return;
    int e = idx >> 3, hh = idx & 7;
    int r = row[e], c = col[e];
    float qk = Q[r * A0 + hh] * K[c * A0 + hh];
    atomicMax((unsigned int*)&m[c * A0 + hh], __float_as_uint(qk));
}

// one thread per (edge, float4 of units): 4x fewer index/score computations,
// b128 gather of V, same (unavoidable) f32 atomic count into L2-resident acc.
__global__ __launch_bounds__(256)
void gat_edge0_agg_kernel(const int* __restrict__ row, const int* __restrict__ col,
                          const float* __restrict__ Q, const float* __restrict__ K,
                          const float* __restrict__ m, const float* __restrict__ V,
                          float* __restrict__ z, float* __restrict__ acc,
                          int total /* E*16 */)
{
    int idx = blockIdx.x * blockDim.x + threadIdx.x;
    if (idx >= total) return;
    int e = idx >> 4, u = (idx & 15) << 2, hh = u >> 3;
    int r = row[e], c = col[e];
    float qk = Q[r * A0 + hh] * K[c * A0 + hh];
    float ee = __expf(qk - m[c * A0 + hh]);
    if ((u & 7) == 0) atomicAdd(&z[c * A0 + hh], ee);
    float4 v4 = *(const float4*)(V + (size_t)r * U0 + u);
    float* ap = acc + (size_t)c * U0 + u;
    atomicAdd(ap + 0, ee * v4.x);
    atomicAdd(ap + 1, ee * v4.y);
    atomicAdd(ap + 2, ee * v4.z);
    atomicAdd(ap + 3, ee * v4.w);
}

__global__ __launch_bounds__(256)
void gat_fin0_kernel(const float* __restrict__ Q, const float* __restrict__ K,
                     const float* __restrict__ m, const float* __restrict__ z,
                     const float* __restrict__ V, const float* __restrict__ b,
                     float* __restrict__ acc /* in-place -> h */, int total /* n*16 */)
{
    int idx = blockIdx.x * blockDim.x + threadIdx.x;
    if (idx >= total) return;
    int i = idx >> 4, u = (idx & 15) << 2, hh = u >> 3;
    float qk = Q[i * A0 + hh] * K[i * A0 + hh];
    float es = __expf(qk - m[i * A0 + hh]);      // self-loop weight
    float rz = 1.f / (z[i * A0 + hh] + es);
    float4 a4 = *(const float4*)(acc + (size_t)i * U0 + u);
    float4 v4 = *(const float4*)(V   + (size_t)i * U0 + u);
    float4 b4 = *(const float4*)(b + u);
    a4.x = fmaxf((a4.x + es * v4.x) * rz + b4.x, 0.f);
    a4.y = fmaxf((a4.y + es * v4.y) * rz + b4.y, 0.f);
    a4.z = fmaxf((a4.z + es * v4.z) * rz + b4.z, 0.f);
    a4.w = fmaxf((a4.w + es * v4.w) * rz + b4.w, 0.f);
    *(float4*)(acc + (size_t)i * U0 + u) = a4;
}

// ---------------------------------------------------------------------------
// Layer 1: single head, attention dim 1.
// ---------------------------------------------------------------------------
__global__ __launch_bounds__(256)
void gat_qk1_kernel(const float* __restrict__ h,
                    const float* __restrict__ Wq, const float* __restrict__ bq,
                    const float* __restrict__ Wk, const float* __restrict__ bk,
                    float* __restrict__ Q1, float* __restrict__ K1,
                    float* __restrict__ m1, float* __restrict__ z1, int n)
{
    int i = blockIdx.x * blockDim.x + threadIdx.x;
    if (i >= n) return;
    const float* hi = h + (size_t)i * U0;
    float q = bq[0], k = bk[0];
    #pragma unroll 8
    for (int j = 0; j < U0; j++) { q += hi[j] * Wq[j]; k += hi[j] * Wk[j]; }
    q = fmaxf(q, 0.f); k = fmaxf(k, 0.f);
    Q1[i] = q; K1[i] = k;
    m1[i] = q * k;      // self-loop score
    z1[i] = 0.f;
}

__global__ __launch_bounds__(256)
void gat_zero_kernel(float* __restrict__ p, int total)
{
    int idx = blockIdx.x * blockDim.x + threadIdx.x;
    if (idx < total) p[idx] = 0.f;
}

__global__ __launch_bounds__(256)
void gat_edge1_max_kernel(const int* __restrict__ row, const int* __restrict__ col,
                          const float* __restrict__ Q1, const float* __restrict__ K1,
                          float* __restrict__ m1, int E)
{
    int e = blockIdx.x * blockDim.x + threadIdx.x;
    if (e >= E) return;
    float qk = Q1[row[e]] * K1[col[e]];
    atomicMax((unsigned int*)&m1[col[e]], __float_as_uint(qk));
}

__global__ __launch_bounds__(256)
void gat_edge1_agg_kernel(const int* __restrict__ row, const int* __restrict__ col,
                          const float* __restrict__ Q1, const float* __restrict__ K1,
                          const float* __restrict__ m1, const float* __restrict__ V,
                          float* __restrict__ z1, float* __restrict__ out,
                          int total /* E*16 */)
{
    int idx = blockIdx.x * blockDim.x + threadIdx.x;
    if (idx >= total) return;
    int e = idx >> 4, u = (idx & 15) << 2;
    int r = row[e], c = col[e];
    float ee = __expf(Q1[r] * K1[c] - m1[c]);
    if (u == 0) atomicAdd(&z1[c], ee);
    float4 v4 = *(const float4*)(V + (size_t)r * NC + u);
    float* op = out + (size_t)c * NC + u;
    atomicAdd(op + 0, ee * v4.x);
    atomicAdd(op + 1, ee * v4.y);
    atomicAdd(op + 2, ee * v4.z);
    atomicAdd(op + 3, ee * v4.w);
}

__global__ __launch_bounds__(256)
void gat_fin1_kernel(const float* __restrict__ Q1, const float* __restrict__ K1,
                     const float* __restrict__ m1, const float* __restrict__ z1,
                     const float* __restrict__ V, const float* __restrict__ b,
                     float* __restrict__ out, int total /* n*16 */)
{
    int idx = blockIdx.x * blockDim.x + threadIdx.x;
    if (idx >= total) return;
    int i = idx >> 4, u = (idx & 15) << 2;
    float es = __expf(Q1[i] * K1[i] - m1[i]);
    float rz = 1.f / (z1[i] + es);
    float4 o4 = *(const float4*)(out + (size_t)i * NC + u);
    float4 v4 = *(const float4*)(V   + (size_t)i * NC + u);
    float4 b4 = *(const float4*)(b + u);
    o4.x = (o4.x + es * v4.x) * rz + b4.x;
    o4.y = (o4.y + es * v4.y) * rz + b4.y;
    o4.z = (o4.z + es * v4.z) * rz + b4.z;
    o4.w = (o4.w + es * v4.w) * rz + b4.w;
    *(float4*)(out + (size_t)i * NC + u) = o4;
}

// ---------------------------------------------------------------------------
extern "C" void kernel_launch(void* const* d_in, const int* in_sizes, int n_in,
                              void* d_out, int out_size, void* d_ws, size_t ws_size,
                              hipStream_t stream)
{
    const float* x   = (const float*)d_in[0];
    const int*   ei  = (const int*)  d_in[1];
    const float* Wq0 = (const float*)d_in[2];
    const float* bq0 = (const float*)d_in[3];
    const float* Wk0 = (const float*)d_in[4];
    const float* bk0 = (const float*)d_in[5];
    const float* Wv0 = (const float*)d_in[6];
    const float* b0  = (const float*)d_in[7];
    const float* Wq1 = (const float*)d_in[8];
    const float* bq1 = (const float*)d_in[9];
    const float* Wk1 = (const float*)d_in[10];
    const float* bk1 = (const float*)d_in[11];
    const float* Wv1 = (const float*)d_in[12];
    const float* b1  = (const float*)d_in[13];

    const int n = in_sizes[0] / F_IN;
    const int E = in_sizes[1] / 2;
    const int* row = ei;        // edge_index[0] : source
    const int* col = ei + E;    // edge_index[1] : destination

    // workspace layout (floats)
    float* ws  = (float*)d_ws;
    float* Q0  = ws;                 ws += (size_t)n * A0;
    float* K0  = ws;                 ws += (size_t)n * A0;
    float* m0  = ws;                 ws += (size_t)n * A0;
    float* z0  = ws;                 ws += (size_t)n * A0;
    float* V0  = ws;                 ws += (size_t)n * U0;
    float* acc = ws;                 ws += (size_t)n * U0;   // becomes h in-place
    float* V1  = ws;                 ws += (size_t)n * NC;
    float* Q1  = ws;                 ws += (size_t)n;
    float* K1  = ws;                 ws += (size_t)n;
    float* m1  = ws;                 ws += (size_t)n;
    float* z1  = ws;                 ws += (size_t)n;

    float* out = (float*)d_out;

    const int tiles = ceil_div(n, 16);
    const int nu    = n * U0;       // n*64
    const int n16   = n * 16;
    const int e8    = E * A0;
    const int e16   = E * 16;

    // ---- layer 0 ----
    gat_proj0_kernel<<<tiles, 160, 0, stream>>>(x, Wq0, bq0, Wk0, bk0, Wv0,
                                                Q0, K0, V0, n);
    gat_init0_kernel<<<ceil_div(nu, 256), 256, 0, stream>>>(Q0, K0, m0, z0, acc, nu);
    gat_edge0_max_kernel<<<ceil_div(e8, 256), 256, 0, stream>>>(row, col, Q0, K0, m0, e8);
    gat_edge0_agg_kernel<<<ceil_div(e16, 256), 256, 0, stream>>>(row, col, Q0, K0, m0,
                                                                 V0, z0, acc, e16);
    gat_fin0_kernel<<<ceil_div(n16, 256), 256, 0, stream>>>(Q0, K0, m0, z0, V0, b0,
                                                            acc, n16);
    // ---- layer 1 (h == acc) ----
    gat_proj1_kernel<<<tiles, 128, 0, stream>>>(acc, Wv1, V1, n);
    gat_qk1_kernel<<<ceil_div(n, 256), 256, 0, stream>>>(acc, Wq1, bq1, Wk1, bk1,
                                                         Q1, K1, m1, z1, n);
    gat_zero_kernel<<<ceil_div(out_size, 256), 256, 0, stream>>>(out, out_size);
    gat_edge1_max_kernel<<<ceil_div(E, 256), 256, 0, stream>>>(row, col, Q1, K1, m1, E);
    gat_edge1_agg_kernel<<<ceil_div(e16, 256), 256, 0, stream>>>(row, col, Q1, K1, m1,
                                                                 V1, z1, out, e16);
    gat_fin1_kernel<<<ceil_div(n16, 256), 256, 0, stream>>>(Q1, K1, m1, z1, V1, b1,
                                                            out, n16);
}